// LabelSmoothingCE_24386824307043
// MI455X (gfx1250) — compile-verified
//
#include <hip/hip_runtime.h>
#include <hip/hip_bf16.h>

// ---------------------------------------------------------------------------
// Label-smoothing cross-entropy, B=4096 rows x V=50257 classes, fp32.
// Bandwidth-bound: 824 MB @ 23.3 TB/s ~= 35 us floor.
//   * one workgroup per row; row streamed through LDS in 51.2 KB chunks by
//     the CDNA5 Tensor Data Mover (TENSOR_LOAD_TO_LDS + s_wait_tensorcnt),
//     2-deep ping-pong: DMA of chunk k+1 overlaps compute of chunk k,
//   * SINGLE-pass online log-softmax (running max/сum with rescale) so no
//     second sweep is needed and the LDS footprint stays ~102 KB
//     -> 3 workgroups per 320 KB WGP for cross-block DMA/compute overlap,
//   * deterministic two-kernel mean reduction (no float atomics).
// ---------------------------------------------------------------------------

typedef __attribute__((ext_vector_type(4))) unsigned int u32x4;
typedef __attribute__((ext_vector_type(8))) int          i32x8;
typedef __attribute__((ext_vector_type(4))) int          i32x4;

#define TPB     512
#define NWAVES  (TPB / 32)
#define CHUNK   12800                 // floats per TDM chunk (51.2 KB, mult of 4)

#define L2E     1.4426950408889634f   // log2(e)
#define LN2     0.6931471805599453f
#define NEG_BIG (-3.402823466e38f)    // -FLT_MAX: merge-safe neutral for max

__device__ __forceinline__ float wred_sum(float v) {
#pragma unroll
    for (int o = 16; o > 0; o >>= 1) v += __shfl_xor(v, o, 32);
    return v;
}

// online-softmax single-element update: m = running max, s = sum e^(x-m)
__device__ __forceinline__ void oupd(float x, float& m, float& s, float& sy) {
    sy += x;
    if (x > m) {
        s = s * __builtin_exp2f((m - x) * L2E) + 1.f;
        m = x;
    } else {
        s += __builtin_exp2f((x - m) * L2E);
    }
}

// merge two (m, s) online-softmax partials
__device__ __forceinline__ void omerge(float& m, float& s, float om, float os) {
    const float nm = fmaxf(m, om);
    s = s * __builtin_exp2f((m - nm) * L2E) + os * __builtin_exp2f((om - nm) * L2E);
    m = nm;
}

// Issue one TDM load of `len` fp32 elements from global `gaddr` to LDS byte
// offset `lds_off`. Descriptor layout per CDNA5 ISA ch.8 (D# groups 0..3).
// This toolchain's builtin takes 6 args:
//   (u32x4 group0, i32x8 group1, i32x4 group2, i32x4 group3, i32x8 extra, cpol)
__device__ __forceinline__ void tdm_load_chunk(unsigned long long gaddr,
                                               unsigned lds_off,
                                               unsigned len) {
    u32x4 g0;
    g0[0] = 1u;                                        // count=1, user desc
    g0[1] = lds_off;                                   // lds_addr (bytes)
    g0[2] = (unsigned)(gaddr & 0xFFFFFFFFull);         // global_addr[31:0]
    g0[3] = ((unsigned)((gaddr >> 32) & 0x01FFFFFFull))// global_addr[56:32]
            | (2u << 30);                              // type = 2 ("image")

    i32x8 g1;
    g1[0] = (int)(2u << 16);                 // wg_mask=0, data_size=2 (4B)
    g1[1] = (int)((len & 0xFFFFu) << 16);    // tensor_dim0[15:0]
    g1[2] = (int)((len >> 16) | (1u << 16)); // tensor_dim0[31:16], tensor_dim1=1
    g1[3] = (int)((len & 0xFFFFu) << 16);    // tile_dim0 = len (fits 16b)
    g1[4] = 1;                               // tile_dim1 = 1, tile_dim2 = 0
    g1[5] = (int)len;                        // tensor_dim0_stride[31:0]
    g1[6] = 0;                               // stride0 hi / stride1 lo
    g1[7] = 0;

    i32x4 z4;
    z4[0] = 0; z4[1] = 0; z4[2] = 0; z4[3] = 0;  // 2-D tensor: groups 2/3 unused
    i32x8 z8;
    z8[0] = 0; z8[1] = 0; z8[2] = 0; z8[3] = 0;
    z8[4] = 0; z8[5] = 0; z8[6] = 0; z8[7] = 0;

    __builtin_amdgcn_tensor_load_to_lds(g0, g1, z4, z4, z8, 0);
}

__global__ void __launch_bounds__(TPB)
lsce_row_kernel(const float* __restrict__ y_pred,
                const int*   __restrict__ y_label,
                float*       __restrict__ row_out,
                int V, float base, float coef) {
    extern __shared__ __align__(16) float lds[];   // [2*CHUNK | 3*NWAVES scratch]

    const int row  = blockIdx.x;
    const int tid  = threadIdx.x;
    const int wid  = tid >> 5;
    const int lane = tid & 31;

    float* sc_m = lds + 2 * CHUNK;
    float* sc_s = sc_m + NWAVES;
    float* sc_y = sc_s + NWAVES;

    const int nchunks = (V + CHUNK - 1) / CHUNK;
    const unsigned long long ga = (unsigned long long)(uintptr_t)y_pred
                                + (unsigned long long)row * (unsigned long long)V * 4ull;

    // prologue: wave 0 launches the DMA of chunk 0
    if (tid < 32) {
        const int len0 = (V < CHUNK) ? V : CHUNK;
        tdm_load_chunk(ga, 0u, (unsigned)len0);
    }

    float m  = NEG_BIG;   // running max
    float s  = 0.f;       // running sum of e^(x - m)
    float sy = 0.f;       // running sum of x

    for (int k = 0; k < nchunks; ++k) {
        const int off = k * CHUNK;
        const int len = (V - off < CHUNK) ? (V - off) : CHUNK;

        if (tid < 32) {
            if (k + 1 < nchunks) {       // prefetch next chunk into other buffer
                const int off1 = off + CHUNK;
                const int len1 = (V - off1 < CHUNK) ? (V - off1) : CHUNK;
                tdm_load_chunk(ga + (unsigned long long)off1 * 4ull,
                               (unsigned)(((k + 1) & 1) * CHUNK * 4),
                               (unsigned)len1);
                __builtin_amdgcn_s_wait_tensorcnt(1);  // chunk k done (in-order)
            } else {
                __builtin_amdgcn_s_wait_tensorcnt(0);  // last chunk done
            }
        }
        __syncthreads();                 // chunk k visible to all waves

        const float*  buf = lds + (k & 1) * CHUNK;
        const float4* b4  = (const float4*)buf;
        const int     len4 = len >> 2;

        for (int j = tid; j < len4; j += TPB) {
            float4 v = b4[j];
            oupd(v.x, m, s, sy);
            oupd(v.y, m, s, sy);
            oupd(v.z, m, s, sy);
            oupd(v.w, m, s, sy);
        }
        for (int i = (len4 << 2) + tid; i < len; i += TPB)
            oupd(buf[i], m, s, sy);

        __syncthreads();                 // all done reading buf before reuse
    }

    // ---- wave-level merge of (m, s) partials + sum of sy ----
#pragma unroll
    for (int o = 16; o > 0; o >>= 1) {
        const float om = __shfl_xor(m, o, 32);
        const float os = __shfl_xor(s, o, 32);
        const float oy = __shfl_xor(sy, o, 32);
        omerge(m, s, om, os);
        sy += oy;
    }
    if (lane == 0) { sc_m[wid] = m; sc_s[wid] = s; sc_y[wid] = sy; }
    __syncthreads();

    // ---- cross-wave merge in wave 0 ----
    if (wid == 0) {
        float fm = (lane < NWAVES) ? sc_m[lane] : NEG_BIG;
        float fs = (lane < NWAVES) ? sc_s[lane] : 0.f;
        float fy = (lane < NWAVES) ? sc_y[lane] : 0.f;
#pragma unroll
        for (int o = 16; o > 0; o >>= 1) {
            const float om = __shfl_xor(fm, o, 32);
            const float os = __shfl_xor(fs, o, 32);
            const float oy = __shfl_xor(fy, o, 32);
            omerge(fm, fs, om, os);
            fy += oy;
        }
        if (lane == 0) {
            const float lse    = fm + __builtin_log2f(fs) * LN2;
            const int   lbl    = y_label[row];
            const float tgt    = y_pred[(unsigned long long)row * (unsigned long long)V
                                        + (unsigned long long)lbl];
            const float sum_lp = fy - (float)V * lse;
            const float tgt_lp = tgt - lse;
            row_out[row] = -(base * sum_lp + coef * tgt_lp);
        }
    }
}

// Deterministic mean over the 4096 per-row losses (single block, fixed order).
__global__ void __launch_bounds__(1024)
lsce_mean_kernel(const float* __restrict__ row_out, float* __restrict__ out, int n) {
    __shared__ float sc[32];
    float s = 0.f;
    for (int i = threadIdx.x; i < n; i += 1024) s += row_out[i];
    s = wred_sum(s);
    const int wid = threadIdx.x >> 5, lane = threadIdx.x & 31;
    if (lane == 0) sc[wid] = s;
    __syncthreads();
    if (threadIdx.x == 0) {
        float t = 0.f;
#pragma unroll
        for (int k = 0; k < 32; ++k) t += sc[k];
        out[0] = t / (float)n;
    }
}

extern "C" void kernel_launch(void* const* d_in, const int* in_sizes, int n_in,
                              void* d_out, int out_size, void* d_ws, size_t ws_size,
                              hipStream_t stream) {
    const float* y_pred  = (const float*)d_in[0];
    const int*   y_label = (const int*)d_in[1];
    const int    Bn      = in_sizes[1];
    const int    V       = in_sizes[0] / Bn;

    const float smoothing = 0.01f;
    const float base = smoothing / (float)(V - 1);
    const float coef = 1.0f - smoothing - base;

    float* rows = (float*)d_ws;

    const size_t ldsBytes = (size_t)(2 * CHUNK) * 4 + 3 * NWAVES * 4;
    (void)hipFuncSetAttribute(reinterpret_cast<const void*>(lsce_row_kernel),
                              hipFuncAttributeMaxDynamicSharedMemorySize,
                              (int)ldsBytes);

    lsce_row_kernel<<<Bn, TPB, ldsBytes, stream>>>(y_pred, y_label, rows,
                                                   V, base, coef);
    lsce_mean_kernel<<<1, 1024, 0, stream>>>(rows, (float*)d_out, Bn);
}